// NormalizedWeightsVarSizedElementReduce_41575283426054
// MI455X (gfx1250) — compile-verified
//
#include <hip/hip_runtime.h>
#include <math.h>

// ---------------------------------------------------------------------------
// NormalizedWeightsVarSizedElementReduce for MI455X (gfx1250, wave32)
//
// out[S,128] = segment_sum( softmax_seg(E @ w_att) * (E @ W^T) )
//            = ( segment_sum( p_i * E_i ) ) @ W^T          (projection is linear)
//
// Passes:
//   K0 init      : pooled=0, sumexp=0, segmax_key=0
//   K1 scores    : wave-per-row dot(E_i, w_att) + atomic segment max (uint-keyed)
//   K2 sumexp    : atomicAdd exp(score - max) per segment
//   K3 pooling   : pooled[seg] += p_i * E_i   (fp32 atomics, L2-resident 4MB)
//   K4 gemm      : out = pooled @ W^T via V_WMMA_F32_16X16X4_F32 (full fp32)
// ---------------------------------------------------------------------------

typedef float v2f __attribute__((ext_vector_type(2)));
typedef float v8f __attribute__((ext_vector_type(8)));

#define DIM 128  // D_IN == D_OUT == 128

// Order-preserving float -> uint key (works with atomicMax on unsigned).
__device__ __forceinline__ unsigned fkey_enc(float v) {
  unsigned u = __float_as_uint(v);
  return (u & 0x80000000u) ? ~u : (u | 0x80000000u);
}
__device__ __forceinline__ float fkey_dec(unsigned k) {
  return (k & 0x80000000u) ? __uint_as_float(k & 0x7FFFFFFFu)
                           : __uint_as_float(~k);
}

// --------------------------- K0: init --------------------------------------
__global__ void init_kernel(unsigned* __restrict__ segmax_key,
                            float* __restrict__ sumexp,
                            float* __restrict__ pooled, int S) {
  int i = blockIdx.x * blockDim.x + threadIdx.x;
  if (i < S * DIM) pooled[i] = 0.0f;
  if (i < S) {
    segmax_key[i] = 0u;  // smaller than encoding of any real score
    sumexp[i] = 0.0f;
  }
}

// ----------------- K1: scores + segment max (wave per row) ------------------
__global__ void scores_segmax_kernel(const float* __restrict__ E,
                                     const float* __restrict__ w_att,
                                     const int* __restrict__ idx,
                                     float* __restrict__ scores,
                                     unsigned* __restrict__ segmax_key, int N) {
  int wave = (int)((blockIdx.x * blockDim.x + threadIdx.x) >> 5);
  int lane = threadIdx.x & 31;
  if (wave >= N) return;

  const float4* erow = (const float4*)(E + (size_t)wave * DIM);
  const float4* w4   = (const float4*)w_att;
  float4 e = erow[lane];           // coalesced 512B per wave
  float4 w = w4[lane];             // L2/L0-resident
  float v = e.x * w.x + e.y * w.y + e.z * w.z + e.w * w.w;

  // wave32 butterfly reduction
  for (int off = 16; off > 0; off >>= 1) v += __shfl_xor(v, off, 32);

  if (lane == 0) {
    scores[wave] = v;
    atomicMax(&segmax_key[idx[wave]], fkey_enc(v));
  }
}

// --------------------- K2: sum of exp per segment ---------------------------
__global__ void sumexp_kernel(const float* __restrict__ scores,
                              const unsigned* __restrict__ segmax_key,
                              const int* __restrict__ idx,
                              float* __restrict__ sumexp, int N) {
  int i = blockIdx.x * blockDim.x + threadIdx.x;
  if (i >= N) return;
  int s = idx[i];
  float mx = fkey_dec(segmax_key[s]);
  atomicAdd(&sumexp[s], __expf(scores[i] - mx));
}

// --------------- K3: weighted segment-sum pooling (wave per row) ------------
__global__ void pool_kernel(const float* __restrict__ E,
                            const float* __restrict__ scores,
                            const unsigned* __restrict__ segmax_key,
                            const float* __restrict__ sumexp,
                            const int* __restrict__ idx,
                            float* __restrict__ pooled, int N) {
  int wave = (int)((blockIdx.x * blockDim.x + threadIdx.x) >> 5);
  int lane = threadIdx.x & 31;
  if (wave >= N) return;

  int s = idx[wave];
  float mx = fkey_dec(segmax_key[s]);
  float p = __expf(scores[wave] - mx) / sumexp[s];

  const float4* erow = (const float4*)(E + (size_t)wave * DIM);
  float4 e = erow[lane];
  float* dst = pooled + (size_t)s * DIM + lane * 4;
  atomicAdd(dst + 0, p * e.x);
  atomicAdd(dst + 1, p * e.y);
  atomicAdd(dst + 2, p * e.z);
  atomicAdd(dst + 3, p * e.w);
}

// ------------- K4: out = pooled(Sx128) @ W^T(128x128), fp32 WMMA ------------
// One wave per 16x16 output tile. A = pooled tile (16xK), B[k][n] = W[n][k].
__global__ void gemm_wmma_kernel(const float* __restrict__ pooled,
                                 const float* __restrict__ w_out,
                                 float* __restrict__ out, int S) {
  int wave = (int)((blockIdx.x * blockDim.x + threadIdx.x) >> 5);
  int lane = threadIdx.x & 31;
  int mtiles = S >> 4;           // S/16 row tiles
  if (wave >= mtiles * (DIM / 16)) return;   // uniform per wave -> EXEC all-1s

  int tm = (wave >> 3) << 4;     // tile row base (8 N-tiles per M-tile)
  int tn = (wave & 7) << 4;      // tile col base
  int half = lane >> 4;          // 0: K pair {0,1}; 1: K pair {2,3}
  int l = lane & 15;

  // ISA 7.12.2: f32 A 16x4 -> lane l(+16) holds row M=l, VGPR0=K(0|2), VGPR1=K(1|3)
  const float* arow = pooled + (size_t)(tm + l) * DIM;   // A row M = l
  const float* brow = w_out  + (size_t)(tn + l) * DIM;   // B col N = l == W row

  v8f c = {};
  for (int k0 = 0; k0 < DIM; k0 += 4) {
    int ka = k0 + (half << 1);
    v2f a = *(const v2f*)(arow + ka);
    v2f b = *(const v2f*)(brow + ka);
    // 8 args: (neg_a, A, neg_b, B, c_mod, C, reuse_a, reuse_b)
    c = __builtin_amdgcn_wmma_f32_16x16x4_f32(
        false, a, false, b, (short)0, c, false, false);
  }

  // f32 C/D 16x16: VGPR r -> M = r + half*8, N = l
  float* obase = out + (size_t)(tm + (half << 3)) * DIM + tn + l;
  #pragma unroll
  for (int r = 0; r < 8; ++r) obase[(size_t)r * DIM] = c[r];
}

// ---------------------------------------------------------------------------
extern "C" void kernel_launch(void* const* d_in, const int* in_sizes, int n_in,
                              void* d_out, int out_size, void* d_ws, size_t ws_size,
                              hipStream_t stream) {
  const float* E     = (const float*)d_in[0];   // [N,128]
  const float* w_att = (const float*)d_in[1];   // [128]
  const float* w_out = (const float*)d_in[2];   // [128,128]
  const int*   idx   = (const int*)d_in[3];     // [N]
  float* out = (float*)d_out;                   // [S,128]

  const int N = in_sizes[0] / DIM;
  const int S = out_size / DIM;                 // avoid reading device scalar

  // Workspace layout (floats): scores[N] | segmax_key[S] | sumexp[S] | pooled[S*128]
  float*    scores     = (float*)d_ws;
  unsigned* segmax_key = (unsigned*)(scores + N);
  float*    sumexp     = (float*)(segmax_key + S);
  float*    pooled     = sumexp + S;

  // K0: init accumulators (every call: deterministic)
  {
    int total = S * DIM;
    init_kernel<<<(total + 255) / 256, 256, 0, stream>>>(segmax_key, sumexp, pooled, S);
  }
  // K1: scores + segment max (8 rows per 256-thread block, wave per row)
  {
    int blocks = (N + 7) / 8;
    scores_segmax_kernel<<<blocks, 256, 0, stream>>>(E, w_att, idx, scores, segmax_key, N);
  }
  // K2: segment sum of exp
  sumexp_kernel<<<(N + 255) / 256, 256, 0, stream>>>(scores, segmax_key, idx, sumexp, N);
  // K3: weighted pooling
  {
    int blocks = (N + 7) / 8;
    pool_kernel<<<blocks, 256, 0, stream>>>(E, scores, segmax_key, sumexp, idx, pooled, N);
  }
  // K4: tiny fp32 WMMA GEMM: out = pooled @ W^T
  {
    int waves = (S >> 4) * (DIM / 16);          // 4096 for S=8192
    int blocks = (waves + 3) / 4;               // 4 waves (128 threads) per block
    gemm_wmma_kernel<<<blocks, 128, 0, stream>>>(pooled, w_out, out, S);
  }
}